// TrmEncLayer_68822555951789
// MI455X (gfx1250) — compile-verified
//
#include <hip/hip_runtime.h>

typedef __attribute__((ext_vector_type(2))) float v2f;
typedef __attribute__((ext_vector_type(8))) float v8f;
typedef __attribute__((ext_vector_type(4))) int   v4i;

// ---------------------------------------------------------------------------
// Async global -> LDS copy (16 bytes), CDNA5 ASYNCcnt path with sync fallback.
// Builtin signature (from hipcc diagnostic): (v4i addrspace(1)*, v4i addrspace(3)*, imm, imm)
// ---------------------------------------------------------------------------
#if __has_builtin(__builtin_amdgcn_global_load_async_to_lds_b128)
#define HAVE_ASYNC_LDS 1
#define ASYNC_COPY16(gp, lp)                                                   \
  __builtin_amdgcn_global_load_async_to_lds_b128(                              \
      (__attribute__((address_space(1))) v4i*)(gp),                            \
      (__attribute__((address_space(3))) v4i*)(lp), 0, 0)
#else
#define HAVE_ASYNC_LDS 0
#define ASYNC_COPY16(gp, lp)                                                   \
  do { *(float4*)(lp) = *(const float4*)(gp); } while (0)
#endif

__device__ __forceinline__ void async_wait_all() {
#if HAVE_ASYNC_LDS
#if __has_builtin(__builtin_amdgcn_s_wait_asynccnt)
  __builtin_amdgcn_s_wait_asynccnt(0);
#else
  asm volatile("s_wait_asynccnt 0x0" ::: "memory");
#endif
#endif
}

// ---------------------------------------------------------------------------
// C[M x N] = A[M x K] @ W[K x N] + bias  (optional ReLU, optional residual).
// f32 WMMA 16x16x4. Block = 128 threads = 4 wave32s; block tile 64M x 64N,
// each wave owns a 16M x 64N strip (4 accumulator fragments).
// K-chunks of 16 staged in double-buffered LDS via async global->LDS b128.
// Requires rows covered by grid, N % 64 == 0, K % 16 == 0.
// ---------------------------------------------------------------------------
__global__ __launch_bounds__(128) void gemm_wmma_f32(
    const float* __restrict__ A, int lda,
    const float* __restrict__ W, int ldw,      // W is K x N row-major, ldw == N
    const float* __restrict__ bias,
    const float* __restrict__ residual, int ldres,
    float* __restrict__ C, int ldc,
    int K, int relu)
{
  constexpr int KC = 16;
  // Pad A rows to 20 floats (80B: 16B-aligned rows; 20r mod 64 distinct for r<16)
  // Pad B rows to 68 floats (272B: 16B-aligned rows; half-wave offsets conflict-free)
  __shared__ __align__(16) float Al[2][64][20];
  __shared__ __align__(16) float Bl[2][16][68];

  const int t     = threadIdx.x;
  const int lane  = t & 31;
  const int wave  = t >> 5;
  const int l16   = lane & 15;
  const int khalf = lane >> 4;            // 0: K=k,k+1   1: K=k+2,k+3
  const int blockN = blockIdx.x * 64;
  const int blockM = blockIdx.y * 64;

  // Cooperative tile fill: 128 threads x 2 x 16B segments per tile.
  auto load_tiles = [&](int b, int kk) {
    // A tile: 64 rows x 16 floats = 256 16B-segments. seg s: r=s>>2, c=(s&3)*4
    {
      const int s0 = t, s1 = t + 128;
      const int r0 = s0 >> 2, c0 = (s0 & 3) * 4;
      const int r1 = s1 >> 2, c1 = (s1 & 3) * 4;
      ASYNC_COPY16(A + (size_t)(blockM + r0) * lda + kk + c0, &Al[b][r0][c0]);
      ASYNC_COPY16(A + (size_t)(blockM + r1) * lda + kk + c1, &Al[b][r1][c1]);
    }
    // B tile: 16 rows x 64 floats = 256 16B-segments. seg s: kr=s>>4, c=(s&15)*4
    {
      const int s0 = t, s1 = t + 128;
      const int k0 = s0 >> 4, c0 = (s0 & 15) * 4;
      const int k1 = s1 >> 4, c1 = (s1 & 15) * 4;
      ASYNC_COPY16(W + (size_t)(kk + k0) * ldw + blockN + c0, &Bl[b][k0][c0]);
      ASYNC_COPY16(W + (size_t)(kk + k1) * ldw + blockN + c1, &Bl[b][k1][c1]);
    }
  };

  v8f acc[4] = {{}, {}, {}, {}};
  const int arow = wave * 16 + l16;       // A-fragment row within block tile

  const int nchunks = K / KC;
  load_tiles(0, 0);                       // prologue: chunk 0 -> buffer 0
  int buf = 0;
  for (int c = 0; c < nchunks; ++c) {
    async_wait_all();                     // this wave's tile DMA done
    __syncthreads();                      // whole tile visible to all waves
    if (c + 1 < nchunks) load_tiles(buf ^ 1, (c + 1) * KC);  // prefetch ahead

#pragma unroll
    for (int k4 = 0; k4 < KC; k4 += 4) {
      const int krow = k4 + khalf * 2;
      v2f a = *(const v2f*)&Al[buf][arow][krow];       // ds_load_b64
      const float* b0 = &Bl[buf][krow][l16];
      const float* b1 = &Bl[buf][krow + 1][l16];
#pragma unroll
      for (int j = 0; j < 4; ++j) {
        v2f bfrag = { b0[j * 16], b1[j * 16] };
        acc[j] = __builtin_amdgcn_wmma_f32_16x16x4_f32(
            false, a, false, bfrag, (short)0, acc[j], false, false);
      }
    }
    buf ^= 1;
  }

  // Epilogue. C/D layout: VGPR i -> M = 8*khalf + i, N = l16.
  const int rbase = blockM + wave * 16 + khalf * 8;
#pragma unroll
  for (int j = 0; j < 4; ++j) {
    const int col = blockN + j * 16 + l16;
    const float bcol = bias[col];
#pragma unroll
    for (int i = 0; i < 8; ++i) {
      const int r = rbase + i;
      float val = acc[j][i] + bcol;
      if (relu) val = fmaxf(val, 0.0f);
      if (residual) val += residual[(size_t)r * ldres + col];
      C[(size_t)r * ldc + col] = val;
    }
  }
}

// ---------------------------------------------------------------------------
// Fused attention middle. One block per sample n (256 threads).
//   kernel[mo] = q·p[mo] + sum_{m,d} q[d] k[m,d] h[m*768+d, mo]
//   S[m,m']   = v[m]·v[m']          (6x6 Gram, 21 unique dots)
//   t[m']     = sum_m kernel[m] S[m,m']
//   attn[e]   = sum_{m'} t[m'] g[m',e] + sum_m kernel[m] v[m,e]
// Result written into x2 row (n,4,:).
// ---------------------------------------------------------------------------
__global__ __launch_bounds__(256) void attn_kernel(
    const float* __restrict__ q, const float* __restrict__ kbuf,
    const float* __restrict__ vbuf, const float* __restrict__ p,
    const float* __restrict__ h, const float* __restrict__ g,
    float* __restrict__ x2)
{
  const int n = blockIdx.x;
  const int tid = threadIdx.x;
  __shared__ float red[256];
  __shared__ float sK[6];
  __shared__ float sS[21];
  __shared__ float sT[6];

  const float* qn = q + (size_t)n * 768;
  const float* kn = kbuf + (size_t)n * 6 * 768;
  const float* vn = vbuf + (size_t)n * 6 * 768;

  float accK[6] = {0.f, 0.f, 0.f, 0.f, 0.f, 0.f};
  for (int j = tid; j < 6 * 768; j += 256) {           // x_q @ h, h is (4608,6)
    const int d = j % 768;
    const float qk = qn[d] * kn[j];
    const float* hj = h + (size_t)j * 6;
#pragma unroll
    for (int mo = 0; mo < 6; ++mo) accK[mo] += qk * hj[mo];
  }
  for (int d = tid; d < 768; d += 256) {               // q @ p.T, p is (6,768)
    const float qd = qn[d];
#pragma unroll
    for (int mo = 0; mo < 6; ++mo) accK[mo] += qd * p[mo * 768 + d];
  }
  float accS[21];
#pragma unroll
  for (int i = 0; i < 21; ++i) accS[i] = 0.f;
  for (int d = tid; d < 768; d += 256) {               // Gram partials
    float v6[6];
#pragma unroll
    for (int m = 0; m < 6; ++m) v6[m] = vn[m * 768 + d];
    int pp = 0;
#pragma unroll
    for (int a2 = 0; a2 < 6; ++a2)
#pragma unroll
      for (int b2 = a2; b2 < 6; ++b2) accS[pp++] += v6[a2] * v6[b2];
  }

  for (int idx = 0; idx < 27; ++idx) {                 // deterministic reductions
    const float vloc = (idx < 6) ? accK[idx] : accS[idx - 6];
    __syncthreads();
    red[tid] = vloc;
    __syncthreads();
    for (int s = 128; s > 0; s >>= 1) {
      if (tid < s) red[tid] += red[tid + s];
      __syncthreads();
    }
    if (tid == 0) { if (idx < 6) sK[idx] = red[0]; else sS[idx - 6] = red[0]; }
  }
  __syncthreads();

  if (tid == 0) {
    float S[6][6];
    int pp = 0;
    for (int a2 = 0; a2 < 6; ++a2)
      for (int b2 = a2; b2 < 6; ++b2) { S[a2][b2] = sS[pp]; S[b2][a2] = sS[pp]; ++pp; }
    for (int mp = 0; mp < 6; ++mp) {
      float tacc = 0.f;
      for (int m = 0; m < 6; ++m) tacc += sK[m] * S[m][mp];
      sT[mp] = tacc;
    }
  }
  __syncthreads();

  float* outrow = x2 + (size_t)(n * 6 + 4) * 768;
  for (int e = tid; e < 768; e += 256) {
    float val = 0.f;
#pragma unroll
    for (int mp = 0; mp < 6; ++mp) val += sT[mp] * g[mp * 768 + e];
#pragma unroll
    for (int m = 0; m < 6; ++m) val += sK[m] * vn[m * 768 + e];
    outrow[e] = val;
  }
}

// ---------------------------------------------------------------------------
// LayerNorm over last dim (768). One block per row; biased variance (ddof=0).
// ---------------------------------------------------------------------------
__global__ __launch_bounds__(256) void layernorm_kernel(
    const float* __restrict__ res, const float* __restrict__ gamma,
    const float* __restrict__ beta, float* __restrict__ out)
{
  const int row = blockIdx.x;
  const int tid = threadIdx.x;
  const float* r = res + (size_t)row * 768;
  __shared__ float red[256];

  float s = 0.f, s2 = 0.f;
  for (int e = tid; e < 768; e += 256) { const float v = r[e]; s += v; s2 += v * v; }

  red[tid] = s;
  __syncthreads();
  for (int st = 128; st > 0; st >>= 1) { if (tid < st) red[tid] += red[tid + st]; __syncthreads(); }
  const float mean = red[0] * (1.0f / 768.0f);
  __syncthreads();
  red[tid] = s2;
  __syncthreads();
  for (int st = 128; st > 0; st >>= 1) { if (tid < st) red[tid] += red[tid + st]; __syncthreads(); }
  const float var = red[0] * (1.0f / 768.0f) - mean * mean;
  const float inv = rsqrtf(var + 1e-12f);

  for (int e = tid; e < 768; e += 256) {
    const float v = (r[e] - mean) * inv;
    out[(size_t)row * 768 + e] = v * gamma[e] + beta[e];
  }
}

// ---------------------------------------------------------------------------
extern "C" void kernel_launch(void* const* d_in, const int* in_sizes, int n_in,
                              void* d_out, int out_size, void* d_ws, size_t ws_size,
                              hipStream_t stream) {
  (void)in_sizes; (void)n_in; (void)out_size; (void)ws_size;
  const float* x     = (const float*)d_in[0];   // (128, 6, 768)
  const float* Wq    = (const float*)d_in[1];
  const float* bq    = (const float*)d_in[2];
  const float* Wk    = (const float*)d_in[3];
  const float* bk    = (const float*)d_in[4];
  const float* Wv    = (const float*)d_in[5];
  const float* bv    = (const float*)d_in[6];
  const float* p     = (const float*)d_in[7];   // (6, 768)
  const float* h     = (const float*)d_in[8];   // (4608, 6)
  const float* g     = (const float*)d_in[9];   // (6, 768)
  const float* Wf    = (const float*)d_in[10];  // (768, 3072)
  const float* bf    = (const float*)d_in[11];
  const float* Ws    = (const float*)d_in[12];  // (3072, 768)
  const float* bs    = (const float*)d_in[13];
  const float* gamma = (const float*)d_in[14];
  const float* beta  = (const float*)d_in[15];

  float* ws   = (float*)d_ws;
  float* q    = ws;                    // 128*768
  float* kbuf = q    + 128 * 768;      // 768*768
  float* vbuf = kbuf + 768 * 768;      // 768*768
  float* x2   = vbuf + 768 * 768;      // 768*768
  float* hmid = x2   + 768 * 768;      // 768*3072
  float* resb = hmid + 768 * 3072;     // 768*768

  // x2 starts as a copy of x; attn kernel overwrites rows (:,4,:)
  (void)hipMemcpyAsync(x2, x, (size_t)768 * 768 * sizeof(float),
                       hipMemcpyDeviceToDevice, stream);

  const dim3 blk(128);
  // q = x[:,4,:] @ Wq + bq     (A base offset 4*768, lda = 6*768)
  gemm_wmma_f32<<<dim3(12, 2), blk, 0, stream>>>(
      x + 4 * 768, 6 * 768, Wq, 768, bq, nullptr, 0, q, 768, 768, 0);
  // k = x @ Wk + bk ; v = x @ Wv + bv   (x viewed as (768, 768))
  gemm_wmma_f32<<<dim3(12, 12), blk, 0, stream>>>(
      x, 768, Wk, 768, bk, nullptr, 0, kbuf, 768, 768, 0);
  gemm_wmma_f32<<<dim3(12, 12), blk, 0, stream>>>(
      x, 768, Wv, 768, bv, nullptr, 0, vbuf, 768, 768, 0);
  // attention middle -> writes x2[:,4,:]
  attn_kernel<<<128, 256, 0, stream>>>(q, kbuf, vbuf, p, h, g, x2);
  // hmid = relu(x2 @ Wf + bf)
  gemm_wmma_f32<<<dim3(48, 12), blk, 0, stream>>>(
      x2, 768, Wf, 3072, bf, nullptr, 0, hmid, 3072, 768, 1);
  // resb = hmid @ Ws + bs + x2
  gemm_wmma_f32<<<dim3(12, 12), blk, 0, stream>>>(
      hmid, 3072, Ws, 768, bs, x2, 768, resb, 768, 3072, 0);
  // layernorm -> d_out
  layernorm_kernel<<<768, 256, 0, stream>>>(resb, gamma, beta, (float*)d_out);
}